// SLMPConv_15169824489703
// MI455X (gfx1250) — compile-verified
//
#include <hip/hip_runtime.h>
#include <hip/hip_bf16.h>

typedef __attribute__((ext_vector_type(2))) float v2f;
typedef __attribute__((ext_vector_type(8))) float v8f;

#define D_FEAT 128
#define EPSN   1e-12f
#define ALPHA_C 1.0f

// ---- monotone float<->uint encoding for atomic scatter-max ----------------
__device__ __forceinline__ unsigned flip_enc(float f) {
    unsigned u = __float_as_uint(f);
    return (u & 0x80000000u) ? ~u : (u | 0x80000000u);
}
__device__ __forceinline__ float flip_dec(unsigned u) {
    unsigned b = (u & 0x80000000u) ? (u ^ 0x80000000u) : ~u;
    // non-finite (e.g. -inf sentinel from empty segments) -> 0, like reference
    if ((b & 0x7F800000u) == 0x7F800000u) return 0.0f;
    return __uint_as_float(b);
}
// flip_enc(-inf) == 0x007FFFFF  (init sentinel)

// ---- init workspace --------------------------------------------------------
__global__ void init_ws_kernel(float* __restrict__ deg, unsigned* __restrict__ agg,
                               int n_nodes) {
    long long i = (long long)blockIdx.x * blockDim.x + threadIdx.x;
    long long tot = (long long)n_nodes * D_FEAT;
    if (i < tot)      agg[i] = 0x007FFFFFu;
    if (i < n_nodes)  deg[i] = 0.0f;
}

// ---- out-degree ------------------------------------------------------------
__global__ void degree_kernel(const int* __restrict__ src, float* __restrict__ deg,
                              int n_edges) {
    int e = blockIdx.x * blockDim.x + threadIdx.x;
    if (e < n_edges) atomicAdd(&deg[src[e]], 1.0f);
}

// ---- h = normalize( (x @ W2^T) * (deg+1)^-1/2 ) via WMMA f32 16x16x4 -------
// block = 256 threads (8 waves). Block handles 16 node rows; wave w handles
// output columns [16w, 16w+16). K-loop: 32 chained v_wmma_f32_16x16x4_f32.
__global__ void __launch_bounds__(256)
gemm_norm_kernel(const float* __restrict__ x, const float* __restrict__ W2,
                 const float* __restrict__ deg, float* __restrict__ h,
                 int n_nodes) {
    __shared__ float lx[16 * D_FEAT];
    __shared__ float ssq[16];

    const int tid  = threadIdx.x;
    const int n0   = blockIdx.x * 16;

    // stage 16x128 tile of x into LDS (clamp rows for safety; N%16==0 here)
    for (int i = tid; i < 16 * D_FEAT; i += 256) {
        int row = i / D_FEAT, col = i % D_FEAT;
        int r = min(n0 + row, n_nodes - 1);
        lx[i] = x[(long long)r * D_FEAT + col];
    }
    if (tid < 16) ssq[tid] = 0.0f;
    __syncthreads();

    const int lane = tid & 31;
    const int o0   = (tid >> 5) * 16;   // this wave's output column block
    const int nm   = lane & 15;         // M index for A frag, N index for B frag
    const int koff = (lane >> 4) * 2;   // K pair: lanes 0-15 -> K0,K1; 16-31 -> K2,K3

    v8f c = {0.f,0.f,0.f,0.f,0.f,0.f,0.f,0.f};
#pragma unroll
    for (int k = 0; k < D_FEAT; k += 4) {
        v2f a, b;
        a.x = lx[nm * D_FEAT + k + koff];
        a.y = lx[nm * D_FEAT + k + koff + 1];
        // B[k',n] = W2[o0+n, k']  (row-major [o,d])
        b.x = W2[(long long)(o0 + nm) * D_FEAT + k + koff];
        b.y = W2[(long long)(o0 + nm) * D_FEAT + k + koff + 1];
        c = __builtin_amdgcn_wmma_f32_16x16x4_f32(false, a, false, b,
                                                  (short)0, c, false, false);
    }

    // row sum-of-squares: reduce over 16 columns per half-wave, add to LDS
#pragma unroll
    for (int r = 0; r < 8; ++r) {
        float p = c[r] * c[r];
        p += __shfl_xor(p, 1, 16);
        p += __shfl_xor(p, 2, 16);
        p += __shfl_xor(p, 4, 16);
        p += __shfl_xor(p, 8, 16);
        int row = r + ((lane < 16) ? 0 : 8);
        if (nm == 0) atomicAdd(&ssq[row], p);
    }
    __syncthreads();

    // scale + write h  (deg-scale first, then L2 normalize with eps clamp)
#pragma unroll
    for (int r = 0; r < 8; ++r) {
        int row  = r + ((lane < 16) ? 0 : 8);
        int node = n0 + row;
        if (node < n_nodes) {
            float s   = rsqrtf(deg[node] + 1.0f);
            float nrm = s * sqrtf(ssq[row]);
            float scl = s / fmaxf(nrm, EPSN);
            h[(long long)node * D_FEAT + o0 + nm] = c[r] * scl;
        }
    }
}

// ---- edge gather + scatter-max: one wave per edge --------------------------
__global__ void __launch_bounds__(256)
scatter_max_kernel(const int* __restrict__ src, const int* __restrict__ dst,
                   const float* __restrict__ h, unsigned* __restrict__ agg,
                   int n_edges) {
    int e    = blockIdx.x * (blockDim.x >> 5) + (threadIdx.x >> 5);
    int lane = threadIdx.x & 31;
    if (e >= n_edges) return;
    int s = src[e], d = dst[e];
    const float4* hp = (const float4*)(h + (long long)s * D_FEAT);
    float4 v = hp[lane];                          // 128B/lane-group, coalesced
    unsigned* ap = agg + (long long)d * D_FEAT + lane * 4;
    atomicMax(ap + 0, flip_enc(v.x));
    atomicMax(ap + 1, flip_enc(v.y));
    atomicMax(ap + 2, flip_enc(v.z));
    atomicMax(ap + 3, flip_enc(v.w));
}

// ---- out = ALPHA * (x @ W1^T) + decode(agg) --------------------------------
__global__ void __launch_bounds__(256)
gemm_out_kernel(const float* __restrict__ x, const float* __restrict__ W1,
                const unsigned* __restrict__ agg, float* __restrict__ out,
                int n_nodes) {
    __shared__ float lx[16 * D_FEAT];

    const int tid = threadIdx.x;
    const int n0  = blockIdx.x * 16;

    for (int i = tid; i < 16 * D_FEAT; i += 256) {
        int row = i / D_FEAT, col = i % D_FEAT;
        int r = min(n0 + row, n_nodes - 1);
        lx[i] = x[(long long)r * D_FEAT + col];
    }
    __syncthreads();

    const int lane = tid & 31;
    const int o0   = (tid >> 5) * 16;
    const int nm   = lane & 15;
    const int koff = (lane >> 4) * 2;

    v8f c = {0.f,0.f,0.f,0.f,0.f,0.f,0.f,0.f};
#pragma unroll
    for (int k = 0; k < D_FEAT; k += 4) {
        v2f a, b;
        a.x = lx[nm * D_FEAT + k + koff];
        a.y = lx[nm * D_FEAT + k + koff + 1];
        b.x = W1[(long long)(o0 + nm) * D_FEAT + k + koff];
        b.y = W1[(long long)(o0 + nm) * D_FEAT + k + koff + 1];
        c = __builtin_amdgcn_wmma_f32_16x16x4_f32(false, a, false, b,
                                                  (short)0, c, false, false);
    }

#pragma unroll
    for (int r = 0; r < 8; ++r) {
        int row  = r + ((lane < 16) ? 0 : 8);
        int node = n0 + row;
        if (node < n_nodes) {
            long long idx = (long long)node * D_FEAT + o0 + nm;
            out[idx] = ALPHA_C * c[r] + flip_dec(agg[idx]);
        }
    }
}

extern "C" void kernel_launch(void* const* d_in, const int* in_sizes, int n_in,
                              void* d_out, int out_size, void* d_ws, size_t ws_size,
                              hipStream_t stream) {
    const float* x  = (const float*)d_in[0];
    const float* W1 = (const float*)d_in[1];
    const float* W2 = (const float*)d_in[2];
    const int*   ei = (const int*)d_in[3];

    const int n_nodes = in_sizes[0] / D_FEAT;
    const int n_edges = in_sizes[3] / 2;
    const int* src = ei;
    const int* dst = ei + n_edges;
    float* out = (float*)d_out;

    // workspace: deg[N] f32 | h[N*128] f32 | agg[N*128] u32  (~51.4 MB)
    char* ws = (char*)d_ws;
    float*    deg = (float*)ws;
    float*    h   = (float*)(ws + sizeof(float) * (size_t)n_nodes);
    unsigned* agg = (unsigned*)(ws + sizeof(float) * ((size_t)n_nodes +
                                 (size_t)n_nodes * D_FEAT));

    long long tot = (long long)n_nodes * D_FEAT;
    int ib = (int)((tot + 255) / 256);
    init_ws_kernel<<<ib, 256, 0, stream>>>(deg, agg, n_nodes);

    degree_kernel<<<(n_edges + 255) / 256, 256, 0, stream>>>(src, deg, n_edges);

    int gblocks = (n_nodes + 15) / 16;
    gemm_norm_kernel<<<gblocks, 256, 0, stream>>>(x, W2, deg, h, n_nodes);

    // 8 waves (edges) per 256-thread block
    scatter_max_kernel<<<(n_edges + 7) / 8, 256, 0, stream>>>(src, dst, h, agg, n_edges);

    gemm_out_kernel<<<gblocks, 256, 0, stream>>>(x, W1, agg, out, n_nodes);
}